// MarkowitzPortfolioOptimizer_14585708937863
// MI455X (gfx1250) — compile-verified
//
#include <hip/hip_runtime.h>
#include <hip/hip_bf16.h>

typedef __attribute__((ext_vector_type(2))) float v2f;
typedef __attribute__((ext_vector_type(8))) float v8f;

#define B_ROWS 16384
#define NASSET 64
#define NITER 200
#define LDS_STRIDE 68   // 64 + 4 pad: (a*68 + k) mod 64 spreads the 16 A-rows over all banks

// ---------------------------------------------------------------------------
// Kernel 1: tiny MLP, one row per thread, weights staged in LDS (broadcast reads)
// ---------------------------------------------------------------------------
__global__ __launch_bounds__(256) void mlp_kernel(
    const float* __restrict__ x, const float* __restrict__ W1, const float* __restrict__ b1,
    const float* __restrict__ W2, const float* __restrict__ b2,
    const float* __restrict__ W3, const float* __restrict__ b3,
    float* __restrict__ muOut)
{
    __shared__ float sW1[128*32];
    __shared__ float sW2[32*16];
    __shared__ float sW3[16*64];
    __shared__ float sb1[32], sb2[16], sb3[64];
    int t = threadIdx.x;
    for (int i = t; i < 128*32; i += 256) sW1[i] = W1[i];
    for (int i = t; i < 32*16;  i += 256) sW2[i] = W2[i];
    for (int i = t; i < 16*64;  i += 256) sW3[i] = W3[i];
    if (t < 32) sb1[t] = b1[t];
    if (t < 16) sb2[t] = b2[t];
    if (t < 64) sb3[t] = b3[t];
    __syncthreads();

    int row = blockIdx.x * 256 + t;

    float h1[32];
#pragma unroll
    for (int j = 0; j < 32; ++j) h1[j] = sb1[j];
    for (int d = 0; d < 128; d += 4) {
        float4 xv = *(const float4*)(x + (size_t)row*128 + d);
#pragma unroll
        for (int j = 0; j < 32; ++j) {
            h1[j] += xv.x * sW1[(d+0)*32+j] + xv.y * sW1[(d+1)*32+j]
                   + xv.z * sW1[(d+2)*32+j] + xv.w * sW1[(d+3)*32+j];
        }
    }
#pragma unroll
    for (int j = 0; j < 32; ++j) h1[j] = fmaxf(h1[j], 0.0f);

    float h2[16];
#pragma unroll
    for (int k = 0; k < 16; ++k) {
        float a = sb2[k];
#pragma unroll
        for (int j = 0; j < 32; ++j) a += h1[j] * sW2[j*16+k];
        h2[k] = fmaxf(a, 0.0f);
    }

    for (int n = 0; n < 64; n += 4) {
        float a0 = sb3[n], a1 = sb3[n+1], a2 = sb3[n+2], a3 = sb3[n+3];
#pragma unroll
        for (int k = 0; k < 16; ++k) {
            float hk = h2[k];
            a0 += hk * sW3[k*64+n+0];
            a1 += hk * sW3[k*64+n+1];
            a2 += hk * sW3[k*64+n+2];
            a3 += hk * sW3[k*64+n+3];
        }
        float4 m; m.x=a0; m.y=a1; m.z=a2; m.w=a3;
        *(float4*)(muOut + (size_t)row*64 + n) = m;
    }
}

// ---------------------------------------------------------------------------
// Kernel 2: power iteration for L = lambda_max(Sigma); writes step = 1/L to ws
// ---------------------------------------------------------------------------
__global__ void powiter_kernel(const float* __restrict__ sigma, float* __restrict__ stepOut)
{
    __shared__ float sS[64*64];
    __shared__ float sv[64];
    __shared__ float red[64];
    int t = threadIdx.x;   // 64 threads
    for (int i = t; i < 64*64; i += 64) sS[i] = sigma[i];
    sv[t] = 0.125f;        // unit-norm start (64 * (1/8)^2 = 1)
    __syncthreads();
    float lam = 1.0f;
    for (int it = 0; it < 96; ++it) {
        float u = 0.0f;
        for (int k = 0; k < 64; ++k) u += sS[k*64 + t] * sv[k];  // symmetric -> coalesced banks
        red[t] = u * u;
        __syncthreads();
        for (int off = 32; off > 0; off >>= 1) {
            if (t < off) red[t] += red[t + off];
            __syncthreads();
        }
        float nrm = sqrtf(red[0]);   // ||Sigma v|| -> lambda_max since v unit
        lam = nrm;
        __syncthreads();
        sv[t] = u / nrm;
        __syncthreads();
    }
    if (t == 0) stepOut[0] = 1.0f / (lam * 1.000001f);
}

// ---------------------------------------------------------------------------
// Kernel 3: FISTA. 256 threads = 8 waves, each wave owns 16 rows.
// State layout == WMMA f32 16x16 D layout of grad^T = Sigma x y^T:
//   lane L<16 : row = base+L,    assets 16t..16t+7   in regs [t][0..7]
//   lane L>=16: row = base+L-16, assets 16t+8..16t+15 in regs [t][0..7]
// ---------------------------------------------------------------------------
__global__ __launch_bounds__(256) void fista_kernel(
    const float* __restrict__ sigma, const float* __restrict__ mu,
    const float* __restrict__ gammaPtr, const float* __restrict__ stepPtr,
    float* __restrict__ wOut)
{
    __shared__ float sS[64*LDS_STRIDE];
    int tid = threadIdx.x;
    for (int i = tid; i < 64*64; i += 256)
        sS[(i >> 6)*LDS_STRIDE + (i & 63)] = sigma[i];
    float step = stepPtr[0];
    float g    = gammaPtr[0];
    __syncthreads();

    const int lane  = tid & 31;
    const int laneM = lane & 15;
    const int kp    = lane >> 4;
    const int wave  = tid >> 5;
    const int row   = blockIdx.x*128 + wave*16 + laneM;

    v8f p[4], w[4], y[4];
#pragma unroll
    for (int t = 0; t < 4; ++t) {
        const float4* src = (const float4*)(mu + (size_t)row*64 + 16*t + 8*kp);
        float4 l4 = src[0];
        float4 h4 = src[1];
        p[t][0] = -g*l4.x; p[t][1] = -g*l4.y; p[t][2] = -g*l4.z; p[t][3] = -g*l4.w;
        p[t][4] = -g*h4.x; p[t][5] = -g*h4.y; p[t][6] = -g*h4.z; p[t][7] = -g*h4.w;
#pragma unroll
        for (int j = 0; j < 8; ++j) { w[t][j] = 1.0f/64.0f; y[t][j] = 1.0f/64.0f; }
    }

    float tk = 1.0f;
    for (int it = 0; it < NITER; ++it) {
        // ---- grad^T = Sigma * y^T + p  via 64x V_WMMA_F32_16X16X4_F32 ----
        v8f acc[4];
#pragma unroll
        for (int t = 0; t < 4; ++t) acc[t] = p[t];

#pragma unroll
        for (int c = 0; c < 16; ++c) {               // contraction chunk: k-assets 4c..4c+3
            const int tt = c >> 2;
            const int m  = (c & 3) * 4;
            float b0, b1;
            if (m < 8) {                             // k-assets held by kp==0 half of pair
                float s2 = __shfl_xor(y[tt][m+2], 16, 32);
                float s3 = __shfl_xor(y[tt][m+3], 16, 32);
                b0 = (kp == 0) ? y[tt][m+0] : s2;
                b1 = (kp == 0) ? y[tt][m+1] : s3;
            } else {                                 // k-assets held by kp==1 half of pair
                const int j = m - 8;
                float s0 = __shfl_xor(y[tt][j+0], 16, 32);
                float s1 = __shfl_xor(y[tt][j+1], 16, 32);
                b0 = (kp == 0) ? s0 : y[tt][j+2];
                b1 = (kp == 0) ? s1 : y[tt][j+3];
            }
            v2f bv; bv[0] = b0; bv[1] = b1;          // B 4x16: lane holds K=2kp, 2kp+1
            const int kbase = 4*c + 2*kp;
#pragma unroll
            for (int mt = 0; mt < 4; ++mt) {
                // A 16x4: lane holds Sigma[16*mt+laneM][kbase], [kbase+1] (b64, bank-clean)
                v2f av = *(const v2f*)&sS[(16*mt + laneM)*LDS_STRIDE + kbase];
                acc[mt] = __builtin_amdgcn_wmma_f32_16x16x4_f32(
                    false, av, false, bv, (short)0, acc[mt], false, false);
            }
        }

        // ---- v = y - step * grad ----
        v8f vv[4];
#pragma unroll
        for (int t = 0; t < 4; ++t)
#pragma unroll
            for (int j = 0; j < 8; ++j) vv[t][j] = y[t][j] - step*acc[t][j];

        // ---- simplex projection: bisection on theta + exact active-set finish ----
        float mx = -3.0e38f, sm = 0.0f;
#pragma unroll
        for (int t = 0; t < 4; ++t)
#pragma unroll
            for (int j = 0; j < 8; ++j) { float z = vv[t][j]; mx = fmaxf(mx, z); sm += z; }
        mx = fmaxf(mx, __shfl_xor(mx, 16, 32));
        sm += __shfl_xor(sm, 16, 32);

        float lo = (sm - 1.0f) * (1.0f/64.0f);       // f(lo) >= 0
        float hi = mx;                               // f(hi) = -1 < 0
        for (int b = 0; b < 26; ++b) {
            float th = 0.5f*(lo + hi);
            float s = 0.0f;
#pragma unroll
            for (int t = 0; t < 4; ++t)
#pragma unroll
                for (int j = 0; j < 8; ++j) s += fmaxf(vv[t][j] - th, 0.0f);
            s += __shfl_xor(s, 16, 32);
            bool gt = (s > 1.0f);
            lo = gt ? th : lo;
            hi = gt ? hi : th;
        }
        float th = 0.5f*(lo + hi);
        float cnt = 0.0f, ss = 0.0f;
#pragma unroll
        for (int t = 0; t < 4; ++t)
#pragma unroll
            for (int j = 0; j < 8; ++j) {
                float z = vv[t][j];
                float m1 = (z > th) ? 1.0f : 0.0f;
                cnt += m1; ss += m1*z;
            }
        cnt += __shfl_xor(cnt, 16, 32);
        ss  += __shfl_xor(ss, 16, 32);
        float theta = (ss - 1.0f) / cnt;             // exact Duchi threshold for this active set

        // ---- FISTA momentum ----
        float tn   = 0.5f*(1.0f + sqrtf(1.0f + 4.0f*tk*tk));
        float coef = (tk - 1.0f)/tn;
#pragma unroll
        for (int t = 0; t < 4; ++t)
#pragma unroll
            for (int j = 0; j < 8; ++j) {
                float wn = fmaxf(vv[t][j] - theta, 0.0f);
                y[t][j] = wn + coef*(wn - w[t][j]);
                w[t][j] = wn;
            }
        tk = tn;
    }

#pragma unroll
    for (int t = 0; t < 4; ++t) {
        float4 o0, o1;
        o0.x=w[t][0]; o0.y=w[t][1]; o0.z=w[t][2]; o0.w=w[t][3];
        o1.x=w[t][4]; o1.y=w[t][5]; o1.z=w[t][6]; o1.w=w[t][7];
        float4* dst = (float4*)(wOut + (size_t)row*64 + 16*t + 8*kp);
        dst[0] = o0; dst[1] = o1;
    }
}

// ---------------------------------------------------------------------------
extern "C" void kernel_launch(void* const* d_in, const int* in_sizes, int n_in,
                              void* d_out, int out_size, void* d_ws, size_t ws_size,
                              hipStream_t stream) {
    const float* x     = (const float*)d_in[0];
    const float* W1    = (const float*)d_in[1];
    const float* b1    = (const float*)d_in[2];
    const float* W2    = (const float*)d_in[3];
    const float* b2    = (const float*)d_in[4];
    const float* W3    = (const float*)d_in[5];
    const float* b3    = (const float*)d_in[6];
    const float* sigma = (const float*)d_in[7];
    const float* gamma = (const float*)d_in[8];

    float* out     = (float*)d_out;
    float* muOut   = out + (size_t)B_ROWS * NASSET;   // second output (mu)
    float* stepBuf = (float*)d_ws;

    mlp_kernel<<<B_ROWS/256, 256, 0, stream>>>(x, W1, b1, W2, b2, W3, b3, muOut);
    powiter_kernel<<<1, 64, 0, stream>>>(sigma, stepBuf);
    fista_kernel<<<B_ROWS/128, 256, 0, stream>>>(sigma, muOut, gamma, stepBuf, out);
}